// MouseBodyModel_25056839205911
// MI455X (gfx1250) — compile-verified
//
#include <hip/hip_runtime.h>
#include <hip/hip_bf16.h>

#define BATCH 512
#define JN    140
#define NB    28
#define NV    13059
#define NC    (3 * BATCH)   // 1536 GEMM columns (b*3 + r)
#define KDIM  (4 * JN)      // 560 GEMM reduction dim (j*4 + n)
#define LDSW  132           // padded LDS row stride (16 rows x 128 cols tile)

typedef __attribute__((ext_vector_type(2))) float v2f;
typedef __attribute__((ext_vector_type(8))) float v8f;

// ---------------------------------------------------------------------------
// Phase 1: per-batch kinematic chain. One thread per batch element.
//   Gchain[(j*12+e)*BATCH + b]          : world transforms (coalesced over b)
//   Gmat[((2j+h)*NC + c)*2 + p]         : G_rel as GEMM B, K-slot pairs packed
//                                         so each lane's two values are one b64
//   Jout[(b*JN+j)*3 + r]                : posed joints (scaled + translated)
// ---------------------------------------------------------------------------
__global__ __launch_bounds__(256)
void mouse_chain_kernel(const float* __restrict__ pose,
                        const float* __restrict__ bone_lengths,
                        const float* __restrict__ cbl,
                        const float* __restrict__ trans,
                        const float* __restrict__ scale,
                        const float* __restrict__ Jt,
                        const int*   __restrict__ parent,
                        const int*   __restrict__ bone_mapper,
                        float* __restrict__ Gmat,
                        float* __restrict__ Gchain,
                        float* __restrict__ Jout)
{
    const int b = blockIdx.x * blockDim.x + threadIdx.x;
    if (b >= BATCH) return;
    const float sc = scale[b];
    const float t0 = trans[b * 3 + 0];
    const float t1 = trans[b * 3 + 1];
    const float t2 = trans[b * 3 + 2];

    for (int i = 0; i < JN; ++i) {
        const float ez = pose[(b * JN + i) * 3 + 0];
        const float ey = pose[(b * JN + i) * 3 + 1];
        const float ex = pose[(b * JN + i) * 3 + 2];
        const float cx = cosf(ex), sx = sinf(ex);
        const float cy = cosf(ey), sy = sinf(ey);
        const float cz = cosf(ez), sz = sinf(ez);
        float R[9];
        R[0] = cz * cy; R[1] = cz * sy * sx - sz * cx; R[2] = cz * sy * cx + sz * sx;
        R[3] = sz * cy; R[4] = sz * sy * sx + cz * cx; R[5] = sz * sy * cx - cz * sx;
        R[6] = -sy;     R[7] = cy * sx;                R[8] = cy * cx;

        float G[12];  // 3 rows of [rot | t]
        if (i == 0) {
            G[0] = R[0]; G[1] = R[1]; G[2]  = R[2]; G[3]  = Jt[0];
            G[4] = R[3]; G[5] = R[4]; G[6]  = R[5]; G[7]  = Jt[1];
            G[8] = R[6]; G[9] = R[7]; G[10] = R[8]; G[11] = Jt[2];
        } else {
            const int p = parent[i];
            float Gp[12];
#pragma unroll
            for (int e = 0; e < 12; ++e) Gp[e] = Gchain[(p * 12 + e) * BATCH + b];
            float s;
            if (i == 1) {
                s = cbl[b];
            } else {
                const int bm = bone_mapper[i];
                s = (bm >= 0) ? 2.0f / (1.0f + expf(-bone_lengths[b * NB + bm] * 0.2f))
                              : 1.0f;
            }
            const float off0 = (Jt[i * 3 + 0] - Jt[p * 3 + 0]) * s;
            const float off1 = (Jt[i * 3 + 1] - Jt[p * 3 + 1]) * s;
            const float off2 = (Jt[i * 3 + 2] - Jt[p * 3 + 2]) * s;
#pragma unroll
            for (int r = 0; r < 3; ++r) {
                const float g0 = Gp[r * 4 + 0], g1 = Gp[r * 4 + 1];
                const float g2 = Gp[r * 4 + 2], g3 = Gp[r * 4 + 3];
                G[r * 4 + 0] = g0 * R[0] + g1 * R[3] + g2 * R[6];
                G[r * 4 + 1] = g0 * R[1] + g1 * R[4] + g2 * R[7];
                G[r * 4 + 2] = g0 * R[2] + g1 * R[5] + g2 * R[8];
                G[r * 4 + 3] = g0 * off0 + g1 * off1 + g2 * off2 + g3;
            }
        }
#pragma unroll
        for (int e = 0; e < 12; ++e) Gchain[(i * 12 + e) * BATCH + b] = G[e];

        // G_rel: translation column becomes t - Rot @ Jt[i]; rotation unchanged.
        const float jx = Jt[i * 3 + 0], jy = Jt[i * 3 + 1], jz = Jt[i * 3 + 2];
#pragma unroll
        for (int r = 0; r < 3; ++r) {
            const float trel = G[r * 4 + 3] -
                (G[r * 4 + 0] * jx + G[r * 4 + 1] * jy + G[r * 4 + 2] * jz);
            const size_t c2 = (size_t)(3 * b + r) * 2;
            Gmat[(size_t)(2 * i    ) * NC * 2 + c2 + 0] = G[r * 4 + 0]; // n=0
            Gmat[(size_t)(2 * i    ) * NC * 2 + c2 + 1] = G[r * 4 + 1]; // n=1
            Gmat[(size_t)(2 * i + 1) * NC * 2 + c2 + 0] = G[r * 4 + 2]; // n=2
            Gmat[(size_t)(2 * i + 1) * NC * 2 + c2 + 1] = trel;         // n=3
        }
        Jout[(b * JN + i) * 3 + 0] = G[3]  * sc + t0;
        Jout[(b * JN + i) * 3 + 1] = G[7]  * sc + t1;
        Jout[(b * JN + i) * 3 + 2] = G[11] * sc + t2;
    }
}

// ---------------------------------------------------------------------------
// Phase 2: fused skinning GEMM via V_WMMA_F32_16X16X4_F32.
//   Out[v, 3b+r] = sum_{j,n} (w[v,j]*vh[v,n]) * G_rel[b,j,r,n]
// Block = 8 waves; wave w owns the 16(M=v) x 16(N) tile at nSub=blockIdx.y*8+w,
// so the block covers a 16 x 128 slab. Two accumulator chains (even/odd j)
// break the WMMA RAW dependency; per 4 wmma: one float4 w-load + four b64
// Gmat loads + 8 v_mul to form fragments. Results staged in LDS, then written
// out address-ordered: per batch column b the block owns 48 contiguous floats
// at (b*NV+v0)*3, emitted by 48 adjacent lanes -> coalesced 192B runs.
// ---------------------------------------------------------------------------
__global__ __launch_bounds__(256)
void mouse_skin_gemm(const float* __restrict__ W,
                     const float* __restrict__ vt,
                     const float* __restrict__ Gmat,
                     const float* __restrict__ scale,
                     const float* __restrict__ trans,
                     float* __restrict__ Vout)
{
    __shared__ float lds[16 * LDSW];

    const int lane  = threadIdx.x & 31;
    const int wave  = threadIdx.x >> 5;
    const int nSub  = blockIdx.y * 8 + wave;   // 0..95 N-subtiles
    const int col   = lane & 15;
    const int khalf = lane >> 4;               // 0: K slots {0,1}  1: {2,3}

    // A-matrix row for this lane (A layout: both lane halves carry M=0..15)
    const int   vRow  = blockIdx.x * 16 + col;
    const bool  vok   = vRow < NV;
    const int   vc    = vok ? vRow : 0;
    const float wmask = vok ? 1.0f : 0.0f;

    // Homogeneous coefficients for this lane's two K slots (n=2*khalf, +1);
    // wmask folded in so padded M rows contribute zero.
    const float c0 = vt[vc * 3 + 2 * khalf] * wmask;              // n in {0,2}
    const float c1 = ((khalf == 0) ? vt[vc * 3 + 1] : 1.0f) * wmask; // n = 1 | 3

    const float* __restrict__ wrow = W + vc * JN;        // 560B rows: 16B aligned
    const int cIdx = nSub * 16 + col;                    // global column 3b+r
    const float* __restrict__ gp =
        Gmat + ((size_t)khalf * NC + cIdx) * 2;          // lane's b64 K-pair

    v8f acc0 = {}, acc1 = {};
#pragma unroll 2
    for (int jq = 0; jq < JN / 4; ++jq) {
        const float4 wq = *(const float4*)(wrow + jq * 4);
        const v2f b0 = *(const v2f*)(gp);
        const v2f b1 = *(const v2f*)(gp + 4 * NC);
        const v2f b2 = *(const v2f*)(gp + 8 * NC);
        const v2f b3 = *(const v2f*)(gp + 12 * NC);
        v2f a;
        a.x = wq.x * c0; a.y = wq.x * c1;
        acc0 = __builtin_amdgcn_wmma_f32_16x16x4_f32(false, a, false, b0,
                                                     (short)0, acc0, false, false);
        a.x = wq.y * c0; a.y = wq.y * c1;
        acc1 = __builtin_amdgcn_wmma_f32_16x16x4_f32(false, a, false, b1,
                                                     (short)0, acc1, false, false);
        a.x = wq.z * c0; a.y = wq.z * c1;
        acc0 = __builtin_amdgcn_wmma_f32_16x16x4_f32(false, a, false, b2,
                                                     (short)0, acc0, false, false);
        a.x = wq.w * c0; a.y = wq.w * c1;
        acc1 = __builtin_amdgcn_wmma_f32_16x16x4_f32(false, a, false, b3,
                                                     (short)0, acc1, false, false);
        gp += 16 * NC;
    }
    const v8f acc = acc0 + acc1;

    // Stage tile in LDS: D layout — VGPR i: lanes 0-15 M=i, lanes 16-31 M=8+i.
#pragma unroll
    for (int i = 0; i < 8; ++i)
        lds[(khalf * 8 + i) * LDSW + wave * 16 + col] = acc[i];
    __syncthreads();

    // Address-ordered writeout. Block slab covers columns [C0, C0+128).
    const int C0     = blockIdx.y * 128;
    const int vr0    = blockIdx.x * 16;
    const int bFirst = C0 / 3;
    const int nGroups = (C0 + 127) / 3 - bFirst + 1;   // batch columns touched
    const int seg = threadIdx.x / 48;                  // 5 segments of 48 lanes
    const int s   = threadIdx.x - seg * 48;            // element within 192B run
    if (seg < 5) {
        const int row = s / 3;                         // vertex row offset 0..15
        const int r   = s - row * 3;                   // xyz component
        const int vr  = vr0 + row;
        for (int g = seg; g < nGroups; g += 5) {
            const int bg = bFirst + g;
            const int lc = 3 * bg + r - C0;            // local column in slab
            if (lc >= 0 && lc < 128 && vr < NV) {
                const float val = lds[row * LDSW + lc];
                Vout[((size_t)bg * NV + vr) * 3 + r] =
                    val * scale[bg] + trans[bg * 3 + r];
            }
        }
    }
}

extern "C" void kernel_launch(void* const* d_in, const int* in_sizes, int n_in,
                              void* d_out, int out_size, void* d_ws, size_t ws_size,
                              hipStream_t stream) {
    const float* pose         = (const float*)d_in[0];
    const float* bone_lengths = (const float*)d_in[1];
    const float* cbl          = (const float*)d_in[2];
    const float* trans        = (const float*)d_in[3];
    const float* scale        = (const float*)d_in[4];
    const float* v_template   = (const float*)d_in[5];
    const float* t_pose       = (const float*)d_in[6];
    const float* weights      = (const float*)d_in[7];
    const int*   parent       = (const int*)d_in[8];
    const int*   bone_mapper  = (const int*)d_in[9];

    float* Gmat   = (float*)d_ws;                    // KDIM*NC floats (3.44 MB)
    float* Gchain = Gmat + (size_t)KDIM * NC;        // JN*12*BATCH floats (3.44 MB)

    float* Vout = (float*)d_out;                     // [BATCH][NV][3]
    float* Jout = Vout + (size_t)BATCH * NV * 3;     // [BATCH][JN][3]

    mouse_chain_kernel<<<dim3((BATCH + 255) / 256), 256, 0, stream>>>(
        pose, bone_lengths, cbl, trans, scale, t_pose, parent, bone_mapper,
        Gmat, Gchain, Jout);

    dim3 grid((NV + 15) / 16, (NC / 16) / 8);        // (817, 12), 8 waves/block
    mouse_skin_gemm<<<grid, 256, 0, stream>>>(
        weights, v_template, Gmat, scale, trans, Vout);
}